// HGNN_conv_76776835383992
// MI455X (gfx1250) — compile-verified
//
#include <hip/hip_runtime.h>

#define N_NODES 100000
#define IN_FT   512
#define OUT_FT  512

#define BLOCK_M 80            // 5 WMMA M-tiles per wave; 100000/80 = 1250 exact
#define M_TILES 5
#define LDS_STRIDE (IN_FT + 4)   // 516: 516 % 64 == 4 -> conflict-free column reads

typedef __attribute__((ext_vector_type(2))) float v2f;
typedef __attribute__((ext_vector_type(8))) float v8f;

// ---------------------------------------------------------------------------
// Kernel 1: h = x @ W + b  via V_WMMA_F32_16X16X4_F32
//   grid (1250, 4), block 256 (8 waves). Block tile 80(M) x 128(N).
//   Each wave owns 16 cols and 5 M-tiles -> each B fragment feeds 5 WMMAs
//   (40 FLOP per byte of W loaded instead of 8). A tile (80x512 fp32) staged
//   in 165 KB of *dynamic* LDS (CDNA5 WGP has 320 KB).
// ---------------------------------------------------------------------------
__global__ __launch_bounds__(256) void gemm_bias_wmma(
    const float* __restrict__ x, const float* __restrict__ w,
    const float* __restrict__ bias, float* __restrict__ h)
{
  extern __shared__ float lds[];

  const int tid   = threadIdx.x;
  const int wave  = tid >> 5;
  const int lane  = tid & 31;
  const int mbase = blockIdx.x * BLOCK_M;
  const int nbase = blockIdx.y * 128 + wave * 16;

  // Cooperative coalesced stage of the 80x512 A tile (b128 loads)
  const float4* __restrict__ xg = (const float4*)(x + (size_t)mbase * IN_FT);
  for (int f = tid; f < BLOCK_M * (IN_FT / 4); f += 256) {
    const int m  = f >> 7;          // / 128
    const int k4 = f & 127;
    float4 v = xg[m * (IN_FT / 4) + k4];
    float* dst = &lds[m * LDS_STRIDE + k4 * 4];
    dst[0] = v.x; dst[1] = v.y; dst[2] = v.z; dst[3] = v.w;
  }
  __syncthreads();

  const int laneLo = lane & 15;
  const int hiSel  = lane >> 4;            // A/B fragment K split at lane 16
  const int col    = nbase + laneLo;

  v8f acc[M_TILES] = {};

  #pragma unroll 2
  for (int k = 0; k < IN_FT; k += 4) {
    const int ke = k + 2 * hiSel;
    v2f bf;                                 // B fragment, shared by 5 WMMAs
    bf.x = w[(size_t)ke * OUT_FT + col];
    bf.y = w[(size_t)(ke + 1) * OUT_FT + col];
    #pragma unroll
    for (int t = 0; t < M_TILES; ++t) {
      const float* ar = &lds[(t * 16 + laneLo) * LDS_STRIDE + ke];
      v2f a; a.x = ar[0]; a.y = ar[1];      // ds_load_b64
      acc[t] = __builtin_amdgcn_wmma_f32_16x16x4_f32(
          false, a, false, bf, (short)0, acc[t], false, false);
    }
  }

  const float bv = bias[col];
  #pragma unroll
  for (int t = 0; t < M_TILES; ++t) {
    float* __restrict__ hp =
        h + (size_t)(mbase + t * 16 + 8 * hiSel) * OUT_FT + col;
    #pragma unroll
    for (int i = 0; i < 8; ++i) hp[(size_t)i * OUT_FT] = acc[t][i] + bv;
  }
}

// ---------------------------------------------------------------------------
// COO -> CSR build (cheap: ~90 MB of traffic vs 6.5 GB of the main phase)
// ---------------------------------------------------------------------------
__global__ __launch_bounds__(256) void hist_rows(
    const int* __restrict__ rows, int* __restrict__ counts, int nnz)
{
  int i = blockIdx.x * 256 + threadIdx.x;
  const int stride = gridDim.x * 256;
  for (; i < nnz; i += stride) atomicAdd(&counts[rows[i]], 1);
}

// Single-workgroup chunked exclusive scan over the 100000 row counts.
__global__ __launch_bounds__(1024) void scan_rowptr(
    const int* __restrict__ counts, int* __restrict__ row_ptr,
    int* __restrict__ cursor)
{
  __shared__ int sdata[1024];
  __shared__ int s_run;
  const int tid = threadIdx.x;
  if (tid == 0) s_run = 0;
  __syncthreads();

  for (int base = 0; base < N_NODES; base += 1024) {
    const int i = base + tid;
    const int v = (i < N_NODES) ? counts[i] : 0;
    const int run = s_run;                  // read before anyone updates it
    sdata[tid] = v;
    __syncthreads();
    for (int off = 1; off < 1024; off <<= 1) {
      const int t = (tid >= off) ? sdata[tid - off] : 0;
      __syncthreads();
      sdata[tid] += t;
      __syncthreads();
    }
    if (i < N_NODES) {
      const int excl = run + sdata[tid] - v;
      row_ptr[i] = excl;
      cursor[i]  = excl;
    }
    __syncthreads();
    if (tid == 0) s_run = run + sdata[1023];
    __syncthreads();
  }
  if (tid == 0) row_ptr[N_NODES] = s_run;
}

__global__ __launch_bounds__(256) void build_csr(
    const int* __restrict__ rows, const int* __restrict__ cols,
    const float* __restrict__ vals, int* __restrict__ cursor,
    int* __restrict__ cols_s, float* __restrict__ vals_s, int nnz)
{
  int i = blockIdx.x * 256 + threadIdx.x;
  const int stride = gridDim.x * 256;
  for (; i < nnz; i += stride) {
    const int r   = rows[i];
    const int pos = atomicAdd(&cursor[r], 1);
    cols_s[pos] = cols[i];
    vals_s[pos] = vals[i];
  }
}

// ---------------------------------------------------------------------------
// Kernel 5: out[row] = sum_e vals[e] * h[cols[e]]  -- wave per row, CSR order.
//   Per edge: 4x global_load_b128 per lane (512 contiguous bytes per wave),
//   16 v_fmac into registers. Four plain b128 stores per lane per row. No fp32 atomics.
//   h rows are ~32x reused -> mostly served from the 192 MB L2.
// ---------------------------------------------------------------------------
__global__ __launch_bounds__(256) void row_gather(
    const int* __restrict__ row_ptr, const int* __restrict__ cols_s,
    const float* __restrict__ vals_s, const float* __restrict__ h,
    float* __restrict__ out)
{
  const int lane = threadIdx.x & 31;
  const int row  = blockIdx.x * 8 + (threadIdx.x >> 5);
  if (row >= N_NODES) return;

  const int beg = row_ptr[row];
  const int end = row_ptr[row + 1];

  float4 a0 = {0,0,0,0}, a1 = {0,0,0,0}, a2 = {0,0,0,0}, a3 = {0,0,0,0};

  for (int e = beg; e < end; ++e) {
    const int   c = cols_s[e];
    const float v = vals_s[e];
    if (e + 1 < end) {                      // cover L2-miss latency of next row
      __builtin_prefetch(h + (size_t)cols_s[e + 1] * OUT_FT + lane * 4, 0, 0);
    }
    const float4* __restrict__ hp = (const float4*)(h + (size_t)c * OUT_FT);
    const float4 h0 = hp[lane];
    const float4 h1 = hp[lane + 32];
    const float4 h2 = hp[lane + 64];
    const float4 h3 = hp[lane + 96];
    a0.x = fmaf(v, h0.x, a0.x); a0.y = fmaf(v, h0.y, a0.y);
    a0.z = fmaf(v, h0.z, a0.z); a0.w = fmaf(v, h0.w, a0.w);
    a1.x = fmaf(v, h1.x, a1.x); a1.y = fmaf(v, h1.y, a1.y);
    a1.z = fmaf(v, h1.z, a1.z); a1.w = fmaf(v, h1.w, a1.w);
    a2.x = fmaf(v, h2.x, a2.x); a2.y = fmaf(v, h2.y, a2.y);
    a2.z = fmaf(v, h2.z, a2.z); a2.w = fmaf(v, h2.w, a2.w);
    a3.x = fmaf(v, h3.x, a3.x); a3.y = fmaf(v, h3.y, a3.y);
    a3.z = fmaf(v, h3.z, a3.z); a3.w = fmaf(v, h3.w, a3.w);
  }

  float4* __restrict__ op = (float4*)(out + (size_t)row * OUT_FT);
  op[lane]      = a0;
  op[lane + 32] = a1;
  op[lane + 64] = a2;
  op[lane + 96] = a3;
}

// ---------------------------------------------------------------------------
// Launch
// ---------------------------------------------------------------------------
extern "C" void kernel_launch(void* const* d_in, const int* in_sizes, int n_in,
                              void* d_out, int out_size, void* d_ws, size_t ws_size,
                              hipStream_t stream)
{
  const float* x      = (const float*)d_in[0];
  const int*   g_rows = (const int*)  d_in[1];
  const int*   g_cols = (const int*)  d_in[2];
  const float* g_vals = (const float*)d_in[3];
  const float* weight = (const float*)d_in[4];
  const float* bias   = (const float*)d_in[5];
  float*       out    = (float*)d_out;

  const int nnz = in_sizes[1];

  // Workspace layout (all 4-byte elements)
  char* ws = (char*)d_ws;
  float* h      = (float*)ws;                 ws += (size_t)N_NODES * OUT_FT * 4;  // 204.8 MB
  int*   cols_s = (int*)ws;                   ws += (size_t)nnz * 4;               // 12.8 MB
  float* vals_s = (float*)ws;                 ws += (size_t)nnz * 4;               // 12.8 MB
  int*   counts = (int*)ws;                   ws += (size_t)N_NODES * 4;
  int*   rowptr = (int*)ws;                   ws += (size_t)(N_NODES + 1) * 4;
  int*   cursor = (int*)ws;

  hipMemsetAsync(counts, 0, (size_t)N_NODES * sizeof(int), stream);

  // Dense transform (overlappable with CSR build by the scheduler)
  dim3 g1(N_NODES / BLOCK_M, OUT_FT / 128);   // (1250, 4)
  gemm_bias_wmma<<<g1, 256, BLOCK_M * LDS_STRIDE * sizeof(float), stream>>>(
      x, weight, bias, h);

  // COO -> CSR
  hist_rows <<<2048, 256, 0, stream>>>(g_rows, counts, nnz);
  scan_rowptr<<<1, 1024, 0, stream>>>(counts, rowptr, cursor);
  build_csr <<<2048, 256, 0, stream>>>(g_rows, g_cols, g_vals, cursor,
                                       cols_s, vals_s, nnz);

  // Gather-accumulate: wave per destination row, no atomics on out
  row_gather<<<(N_NODES + 7) / 8, 256, 0, stream>>>(rowptr, cols_s, vals_s, h, out);
}